// HyperGNNLayer_51118700757120
// MI455X (gfx1250) — compile-verified
//
#include <hip/hip_runtime.h>

typedef float v2f __attribute__((ext_vector_type(2)));
typedef float v8f __attribute__((ext_vector_type(8)));

#define NB   4
#define NN   96
#define FF   16
#define BN   (NB * NN)          // 384 total rows
#define JK   (NN * NN)          // 9216 (j,k) pairs per (b,i)

// ---------------------------------------------------------------------------
// Kernel 1: both 16x16 MLPs via V_WMMA_F32_16X16X4_F32 (exact f32 GEMM).
// One wave (32 threads) per 16-row tile of the flattened (384,16) x.
// A layout (16x4 f32):  lane m = lane&15; vgpr0 = X[m, k0+2*half], vgpr1 = X[m, k0+2*half+1]
// B layout (4x16 f32):  vgpr0 = W[k0+2*half, n=lane&15], vgpr1 = W[k0+2*half+1, n]
// C/D layout (16x16 f32): c[r] = D[r + 8*half, lane&15]
// ---------------------------------------------------------------------------
__device__ __forceinline__ v8f mlp2_tile(const float* __restrict__ xlds,
                                         float* __restrict__ hlds,
                                         const float* __restrict__ w1,
                                         const float* __restrict__ b1,
                                         const float* __restrict__ w2,
                                         const float* __restrict__ b2,
                                         int l16, int hh) {
    // ---- layer 1: relu(X @ w1 + b1) ----
    v8f c = {};
#pragma unroll
    for (int ch = 0; ch < 4; ++ch) {
        const int k0 = ch * 4 + hh * 2;
        v2f a; a.x = xlds[l16 * FF + k0];     a.y = xlds[l16 * FF + k0 + 1];
        v2f b; b.x = w1[k0 * FF + l16];       b.y = w1[(k0 + 1) * FF + l16];
        c = __builtin_amdgcn_wmma_f32_16x16x4_f32(false, a, false, b,
                                                  (short)0, c, false, false);
    }
    const float bv1 = b1[l16];
#pragma unroll
    for (int r = 0; r < 8; ++r) {
        hlds[(r + 8 * hh) * FF + l16] = fmaxf(c[r] + bv1, 0.f);
    }
    __syncthreads();

    // ---- layer 2: relu(H @ w2 + b2) ----
    v8f d = {};
#pragma unroll
    for (int ch = 0; ch < 4; ++ch) {
        const int k0 = ch * 4 + hh * 2;
        v2f a; a.x = hlds[l16 * FF + k0];     a.y = hlds[l16 * FF + k0 + 1];
        v2f b; b.x = w2[k0 * FF + l16];       b.y = w2[(k0 + 1) * FF + l16];
        d = __builtin_amdgcn_wmma_f32_16x16x4_f32(false, a, false, b,
                                                  (short)0, d, false, false);
    }
    const float bv2 = b2[l16];
#pragma unroll
    for (int r = 0; r < 8; ++r) d[r] = fmaxf(d[r] + bv2, 0.f);
    __syncthreads();   // hlds reusable by the next call
    return d;
}

__global__ __launch_bounds__(32)
void mlp_kernel(const float* __restrict__ x,
                const float* __restrict__ nf_w1, const float* __restrict__ nf_b1,
                const float* __restrict__ nf_w2, const float* __restrict__ nf_b2,
                const float* __restrict__ ns_w1, const float* __restrict__ ns_b1,
                const float* __restrict__ ns_w2, const float* __restrict__ ns_b2,
                float* __restrict__ x1_out, float* __restrict__ self_out) {
    __shared__ __align__(16) float xlds[16 * FF];
    __shared__ __align__(16) float hlds[16 * FF];

    const int lane = threadIdx.x;
    const int l16  = lane & 15;
    const int hh   = lane >> 4;
    const int rowbase = blockIdx.x * 16;

    // stage the 16x16 x-tile into LDS (8 floats per lane)
    const float* xt = x + (size_t)rowbase * FF;
#pragma unroll
    for (int i = 0; i < 8; ++i) xlds[lane * 8 + i] = xt[lane * 8 + i];
    __syncthreads();

    v8f x1v = mlp2_tile(xlds, hlds, nf_w1, nf_b1, nf_w2, nf_b2, l16, hh);
    v8f sfv = mlp2_tile(xlds, hlds, ns_w1, ns_b1, ns_w2, ns_b2, l16, hh);

#pragma unroll
    for (int r = 0; r < 8; ++r) {
        const int row = rowbase + r + 8 * hh;
        x1_out[row * FF + l16]   = x1v[r];
        self_out[row * FF + l16] = sfv[r];
    }
}

// ---------------------------------------------------------------------------
// Kernel 2: bandwidth-bound stream of A (14 MB) and W (226 MB, ~half skipped
// via the A==0 predicate). One block per (b,i); 512 threads; each thread owns
// 4 consecutive f channels and strides the 9216 (j,k) pairs with float4 loads
// of W (32 lanes -> 512 contiguous bytes per wave-step).
// ---------------------------------------------------------------------------
__global__ __launch_bounds__(512)
void contract_kernel(const float* __restrict__ A, const float* __restrict__ W,
                     const float* __restrict__ x1, const float* __restrict__ selfo,
                     float* __restrict__ out) {
    __shared__ __align__(16) float  xs[NN * FF];     // x1[b], 6 KB
    __shared__ __align__(16) float4 red4[512];       // partial sums, 8 KB
    __shared__ float redA[128];                      // partial A sums

    const int bi = blockIdx.x;                       // 0..383  (= b*96 + i)
    const int b  = bi / NN;
    const float* __restrict__ Arow = A + (size_t)bi * JK;
    const float* __restrict__ Wrow = W + (size_t)bi * JK * FF;

    const int t = threadIdx.x;
    for (int idx = t; idx < NN * FF; idx += 512) xs[idx] = x1[b * NN * FF + idx];
    __syncthreads();

    const int f4 = t & 3;        // which quad of f channels
    const int g  = t >> 2;       // 0..127, strides the jk space
    float4 s = make_float4(0.f, 0.f, 0.f, 0.f);
    float asum = 0.f;

    for (int jk = g; jk < JK; jk += 128) {
        __builtin_prefetch(Arow + jk + 128 * 8, 0, 0);   // global_prefetch_b8
        const float a = Arow[jk];
        asum += a;
        if (a != 0.f) {
            const int j = jk / NN;
            const int k = jk - j * NN;
            const float4 w  = *(const float4*)(Wrow + (size_t)jk * FF + f4 * 4);
            const float4 xj = *(const float4*)(xs + j * FF + f4 * 4);
            const float4 xk = *(const float4*)(xs + k * FF + f4 * 4);
            s.x += a * w.x * xj.x * xk.x;
            s.y += a * w.y * xj.y * xk.y;
            s.z += a * w.z * xj.z * xk.z;
            s.w += a * w.w * xj.w * xk.w;
        }
    }

    red4[t] = s;
    if (f4 == 0) redA[g] = asum;
    __syncthreads();

    if (t < FF) {
        const int f = t, q = f >> 2, r = f & 3;
        float tot = 0.f, at = 0.f;
        for (int gg = 0; gg < 128; ++gg) {
            const float4 v = red4[gg * 4 + q];
            tot += (r == 0) ? v.x : (r == 1) ? v.y : (r == 2) ? v.z : v.w;
            at  += redA[gg];
        }
        const float res = (at > 0.f) ? (tot / at) : 0.f;
        out[bi * FF + f] = res + selfo[bi * FF + f];
    }
}

// ---------------------------------------------------------------------------
extern "C" void kernel_launch(void* const* d_in, const int* in_sizes, int n_in,
                              void* d_out, int out_size, void* d_ws, size_t ws_size,
                              hipStream_t stream) {
    const float* A     = (const float*)d_in[0];
    const float* W     = (const float*)d_in[1];
    const float* x     = (const float*)d_in[2];
    const float* nf_w1 = (const float*)d_in[3];
    const float* nf_b1 = (const float*)d_in[4];
    const float* nf_w2 = (const float*)d_in[5];
    const float* nf_b2 = (const float*)d_in[6];
    const float* ns_w1 = (const float*)d_in[7];
    const float* ns_b1 = (const float*)d_in[8];
    const float* ns_w2 = (const float*)d_in[9];
    const float* ns_b2 = (const float*)d_in[10];

    float* out   = (float*)d_out;
    float* x1    = (float*)d_ws;           // 384*16 f32
    float* selfo = x1 + BN * FF;           // 384*16 f32  (48 KB total scratch)

    mlp_kernel<<<BN / 16, 32, 0, stream>>>(x, nf_w1, nf_b1, nf_w2, nf_b2,
                                           ns_w1, ns_b1, ns_w2, ns_b2,
                                           x1, selfo);
    contract_kernel<<<BN, 512, 0, stream>>>(A, W, x1, selfo, out);
}